// GATMechanism_5411658793121
// MI455X (gfx1250) — compile-verified
//
#include <hip/hip_runtime.h>
#include <hip/hip_bf16.h>

#define N_NODES 1024
#define F_DIM   128
#define NEG_SLOPE 0.01f

typedef __attribute__((ext_vector_type(2))) float v2f;
typedef __attribute__((ext_vector_type(4))) float v4f;
typedef __attribute__((ext_vector_type(8))) float v8f;

// ---------------------------------------------------------------------------
// Kernel 1: Vp = V @ W^T + bias   via V_WMMA_F32_16X16X4_F32
// One wave (32 lanes) per 16x16 output tile. 64 M-tiles x 8 N-tiles = 512 waves.
// ---------------------------------------------------------------------------
__global__ __launch_bounds__(256) void gat_proj_wmma(
    const float* __restrict__ V,        // (1024,128)
    const float* __restrict__ W,        // (128,128) row n = output feature
    const float* __restrict__ bias,     // (128)
    float* __restrict__ Vp)             // (1024,128)
{
    const int tid    = blockIdx.x * blockDim.x + threadIdx.x;
    const int waveId = tid >> 5;
    const int lane   = tid & 31;
    if (waveId >= (N_NODES / 16) * (F_DIM / 16)) return;

    const int mt = waveId >> 3;         // 0..63
    const int nt = waveId & 7;          // 0..7
    const int m0 = mt * 16;
    const int n0 = nt * 16;

    const int half = lane >> 4;         // 0: K pair {0,1}, 1: K pair {2,3}
    const int lrow = lane & 15;

    const float* aRow = V + (size_t)(m0 + lrow) * F_DIM + 2 * half;
    const float* bRow = W + (size_t)(n0 + lrow) * F_DIM + 2 * half;

    v8f acc = {};
    #pragma unroll
    for (int kb = 0; kb < F_DIM; kb += 4) {
        v2f a = *(const v2f*)(aRow + kb);
        v2f b = *(const v2f*)(bRow + kb);
        acc = __builtin_amdgcn_wmma_f32_16x16x4_f32(
            /*neg_a=*/false, a, /*neg_b=*/false, b,
            /*c_mod=*/(short)0, acc, /*reuse_a=*/false, /*reuse_b=*/false);
    }

    const int ncol = n0 + lrow;
    const float bn = bias[ncol];
    #pragma unroll
    for (int r = 0; r < 8; ++r) {
        const int mrow = m0 + r + 8 * half;
        Vp[(size_t)mrow * F_DIM + ncol] = acc[r] + bn;
    }
}

// ---------------------------------------------------------------------------
// Kernel 2: s_i[m] = Vp[m,:]·a[0:128], s_j[m] = Vp[m,:]·a[128:256]
// ---------------------------------------------------------------------------
__global__ __launch_bounds__(256) void gat_rowdots(
    const float* __restrict__ Vp, const float* __restrict__ a,
    float* __restrict__ sI, float* __restrict__ sJ)
{
    const int m = blockIdx.x * blockDim.x + threadIdx.x;
    if (m >= N_NODES) return;
    const float* row = Vp + (size_t)m * F_DIM;
    float di = 0.f, dj = 0.f;
    #pragma unroll 4
    for (int k = 0; k < F_DIM; ++k) {
        const float v = row[k];
        di = fmaf(v, a[k], di);
        dj = fmaf(v, a[F_DIM + k], dj);
    }
    sI[m] = di;
    sJ[m] = dj;
}

// ---------------------------------------------------------------------------
// Kernel 3: per row i -- score, leaky-relu, softmax over j, H write.
// Block = 256 threads (8 waves), grid = 1024 (one block per i).
// time_enc streamed once (non-temporal), H streamed once (non-temporal).
// ---------------------------------------------------------------------------
__global__ __launch_bounds__(256) void gat_softmax_h(
    const float* __restrict__ Vp,       // (1024,128) hot in L2
    const float* __restrict__ sI,
    const float* __restrict__ sJ,
    const float* __restrict__ E,        // (1024,1024)
    const float* __restrict__ te,       // (1024,1024,128) streamed
    const float* __restrict__ a,        // (384,) ; a_t = a[256:384]
    float* __restrict__ H)              // (1024,1024,128) streamed
{
    __shared__ float zbuf[N_NODES];     // per-j scores -> exp values
    __shared__ float at[F_DIM];
    __shared__ float red[8];
    __shared__ float sMax, sSum;

    const int i    = blockIdx.x;
    const int tid  = threadIdx.x;
    const int wid  = tid >> 5;          // 0..7
    const int lane = tid & 31;

    if (tid < F_DIM) at[tid] = a[2 * F_DIM + tid];
    __syncthreads();

    // Each lane's 4-element slice of a_t, hoisted to registers.
    const float a0 = at[4 * lane + 0];
    const float a1 = at[4 * lane + 1];
    const float a2 = at[4 * lane + 2];
    const float a3 = at[4 * lane + 3];
    const float si = sI[i];

    // ---- Phase 1: scores. Wave w handles j in [w*128, w*128+128). ----
    const size_t rowBase = (size_t)i << 10;   // i * 1024
    for (int t = 0; t < 128; ++t) {
        const int j = wid * 128 + t;
        const v4f* p = (const v4f*)(te + (rowBase + j) * F_DIM) + lane;
        const v4f v = __builtin_nontemporal_load(p);
        float d = v.x * a0 + v.y * a1 + v.z * a2 + v.w * a3;
        // wave32 reduction
        d += __shfl_xor(d, 16);
        d += __shfl_xor(d, 8);
        d += __shfl_xor(d, 4);
        d += __shfl_xor(d, 2);
        d += __shfl_xor(d, 1);
        if (lane == 0) {
            float z = si + sJ[j] + d;
            zbuf[j] = (z > 0.f) ? z : NEG_SLOPE * z;
        }
    }
    __syncthreads();

    // ---- Phase 2a: row max ----
    float m = -3.402823466e+38f;
    #pragma unroll
    for (int q = 0; q < 4; ++q) m = fmaxf(m, zbuf[tid + 256 * q]);
    m = fmaxf(m, __shfl_xor(m, 16));
    m = fmaxf(m, __shfl_xor(m, 8));
    m = fmaxf(m, __shfl_xor(m, 4));
    m = fmaxf(m, __shfl_xor(m, 2));
    m = fmaxf(m, __shfl_xor(m, 1));
    if (lane == 0) red[wid] = m;
    __syncthreads();
    if (tid == 0) {
        float mm = red[0];
        #pragma unroll
        for (int q = 1; q < 8; ++q) mm = fmaxf(mm, red[q]);
        sMax = mm;
    }
    __syncthreads();
    const float maxv = sMax;

    // ---- Phase 2b: exp + sum ----
    float s = 0.f;
    #pragma unroll
    for (int q = 0; q < 4; ++q) {
        const int idx = tid + 256 * q;
        const float e = __expf(zbuf[idx] - maxv);
        zbuf[idx] = e;
        s += e;
    }
    s += __shfl_xor(s, 16);
    s += __shfl_xor(s, 8);
    s += __shfl_xor(s, 4);
    s += __shfl_xor(s, 2);
    s += __shfl_xor(s, 1);
    if (lane == 0) red[wid] = s;
    __syncthreads();
    if (tid == 0) {
        float ss = 0.f;
        #pragma unroll
        for (int q = 0; q < 8; ++q) ss += red[q];
        sSum = ss;
    }
    __syncthreads();
    const float inv = __frcp_rn(sSum);

    // ---- Phase 3: H[i,j,:] = att * E[i,j] * Vp[j,:] ----
    for (int t = 0; t < 128; ++t) {
        const int j = wid * 128 + t;
        const float sc = zbuf[j] * inv * E[rowBase + j];
        const v4f vp = *((const v4f*)(Vp + (size_t)j * F_DIM) + lane);
        v4f h;
        h.x = vp.x * sc; h.y = vp.y * sc; h.z = vp.z * sc; h.w = vp.w * sc;
        v4f* op = (v4f*)(H + (rowBase + j) * F_DIM) + lane;
        __builtin_nontemporal_store(h, op);
    }
}

// ---------------------------------------------------------------------------
extern "C" void kernel_launch(void* const* d_in, const int* in_sizes, int n_in,
                              void* d_out, int out_size, void* d_ws, size_t ws_size,
                              hipStream_t stream) {
    const float* V    = (const float*)d_in[0];   // (1024,128)
    const float* E    = (const float*)d_in[1];   // (1024,1024)
    const float* te   = (const float*)d_in[2];   // (1024,1024,128)
    const float* Ww   = (const float*)d_in[3];   // (128,128)
    const float* Wb   = (const float*)d_in[4];   // (128,)
    const float* a    = (const float*)d_in[5];   // (384,1)
    float* H          = (float*)d_out;           // (1024,1024,128)

    float* ws = (float*)d_ws;
    float* Vp = ws;                              // 1024*128
    float* sI = ws + N_NODES * F_DIM;            // 1024
    float* sJ = sI + N_NODES;                    // 1024

    // 512 waves, 8 waves per block -> 64 blocks
    gat_proj_wmma<<<64, 256, 0, stream>>>(V, Ww, Wb, Vp);
    gat_rowdots<<<N_NODES / 256, 256, 0, stream>>>(Vp, a, sI, sJ);
    gat_softmax_h<<<N_NODES, 256, 0, stream>>>(Vp, sI, sJ, E, te, a, H);
}